// NBVHModel_26173530701858
// MI455X (gfx1250) — compile-verified
//
#include <hip/hip_runtime.h>
#include <stdint.h>

// ---------------------------------------------------------------------------
// NBVH hash-grid encoding for MI455X (gfx1250).
// Memory-bound gather kernel: 128MB NT-stream out + ~512MB gathers from a
// 64MB table (L2-resident, L2=192MB). The 8-corner x 128B row gather per
// (ray,depth) maps 1:1 onto the CDNA5 Tensor Data Mover "gather mode"
// (32-bit indices => up to 8 rows), issued asynchronously and pipelined
// across the 8 depths via TENSORcnt. Trilinear interp is a factored 7-lerp
// tree; box reciprocals are v_rcp_f32 + 1 Newton step, shared by both points.
// ---------------------------------------------------------------------------

#define N_POINTS        2
#define ENC_DIM         32
#define ENC_DEPTH       8
#define TABLE_MASK      0x7FFFFu        // TABLE_SIZE = 524288 = 2^19
#define WAVES_PER_BLOCK 8
#define CORNERS         8

typedef __attribute__((ext_vector_type(4))) unsigned int u32x4;
typedef __attribute__((ext_vector_type(8))) int          i32x8;
typedef __attribute__((ext_vector_type(4))) int          i32x4;

// (idx ^ CM) & 0x7FFFF == idx ^ (CM & 0x7FFFF) since idx < 2^19 and XOR/AND
// are bitwise => precompute the masked 64-bit products CORNERS[c]*PIS[c].
constexpr unsigned CM_LOW[CORNERS] = {
  (unsigned)((1ull * 774363409ull)  & 0x7FFFFull),
  (unsigned)((2ull * 2654435761ull) & 0x7FFFFull),
  (unsigned)((3ull * 805459861ull)  & 0x7FFFFull),
  (unsigned)((4ull * 100000007ull)  & 0x7FFFFull),
  (unsigned)((5ull * 334363391ull)  & 0x7FFFFull),
  (unsigned)((6ull * 1334363413ull) & 0x7FFFFull),
  (unsigned)((7ull * 734363407ull)  & 0x7FFFFull),
  (unsigned)((8ull * 2134363393ull) & 0x7FFFFull),
};

__device__ __forceinline__ void wait_tensorcnt_le(int n) {
  // s_wait_tensorcnt needs an immediate; loop below is fully unrolled so
  // this switch folds to a single instruction per depth.
  switch (n) {
    case 0: __builtin_amdgcn_s_wait_tensorcnt(0); break;
    case 1: __builtin_amdgcn_s_wait_tensorcnt(1); break;
    case 2: __builtin_amdgcn_s_wait_tensorcnt(2); break;
    case 3: __builtin_amdgcn_s_wait_tensorcnt(3); break;
    case 4: __builtin_amdgcn_s_wait_tensorcnt(4); break;
    case 5: __builtin_amdgcn_s_wait_tensorcnt(5); break;
    case 6: __builtin_amdgcn_s_wait_tensorcnt(6); break;
    default: __builtin_amdgcn_s_wait_tensorcnt(7); break;
  }
}

// v_rcp_f32 (~1e-5 rel) + one Newton-Raphson step -> ~1ulp reciprocal.
// ext in [0.5, 1.5): no denormal/overflow concerns.
__device__ __forceinline__ float fast_rcp(float d) {
  float r = __builtin_amdgcn_rcpf(d);
  return r * fmaf(-d, r, 2.0f);
}

__device__ __forceinline__ float clamp01(float v) {
  return fminf(fmaxf(v, 0.0f), 1.0f);
}

__global__ __launch_bounds__(WAVES_PER_BLOCK * 32)
void nbvh_enc_kernel(const float* __restrict__ inp,
                     const int*   __restrict__ history,
                     const float* __restrict__ emb,
                     const float* __restrict__ nodes_min,
                     const float* __restrict__ nodes_extent,
                     float* __restrict__ out,
                     int n_rays)
{
  // Per-wave staging: 8 depths x 8 corners x 32 f32 = 8KB; 8 waves = 64KB.
  __shared__ float smem[WAVES_PER_BLOCK * ENC_DEPTH * CORNERS * ENC_DIM];

  const int lane = (int)(threadIdx.x & 31u);
  const int wave = __builtin_amdgcn_readfirstlane((int)(threadIdx.x >> 5));
  const int ray  = (int)blockIdx.x * WAVES_PER_BLOCK + wave;
  if (ray >= n_rays) return;

  float* wbase = &smem[wave * (ENC_DEPTH * CORNERS * ENC_DIM)];
  // Flat LDS address: addr[31:0] IS the LDS byte offset (aperture spec).
  const unsigned           lds_base = (unsigned)(uintptr_t)wbase;
  const unsigned long long emb_a    = (unsigned long long)(uintptr_t)emb;

  // Ray sample points (wave-uniform -> scalar loads).
  float px[N_POINTS][3];
#pragma unroll
  for (int p = 0; p < N_POINTS; ++p)
#pragma unroll
    for (int k = 0; k < 3; ++k)
      px[p][k] = inp[(size_t)ray * (N_POINTS * 3) + p * 3 + k];

  float nmn[ENC_DEPTH][3], rex[ENC_DEPTH][3];

  // ---------------- Phase 1: issue 8 asynchronous TDM row-gathers ----------
#pragma unroll
  for (int i = 0; i < ENC_DEPTH; ++i) {
    const unsigned idx = (unsigned)history[ray * ENC_DEPTH + i];
#pragma unroll
    for (int k = 0; k < 3; ++k) {
      nmn[i][k] = nodes_min[idx * 3 + k];
      rex[i][k] = fast_rcp(nodes_extent[idx * 3 + k]);  // shared by both points
    }
    const int t0 = (int)((idx ^ CM_LOW[0]) & TABLE_MASK);
    const int t1 = (int)((idx ^ CM_LOW[1]) & TABLE_MASK);
    const int t2 = (int)((idx ^ CM_LOW[2]) & TABLE_MASK);
    const int t3 = (int)((idx ^ CM_LOW[3]) & TABLE_MASK);
    const int t4 = (int)((idx ^ CM_LOW[4]) & TABLE_MASK);
    const int t5 = (int)((idx ^ CM_LOW[5]) & TABLE_MASK);
    const int t6 = (int)((idx ^ CM_LOW[6]) & TABLE_MASK);
    const int t7 = (int)((idx ^ CM_LOW[7]) & TABLE_MASK);

    // D# group 0: count=1 | gather_index_size(32b)<<30 | gather_mode<<31;
    //             lds_addr; global_addr[56:0]; type=2 ("image") in [127:126].
    u32x4 g0;
    g0.x = 0xC0000001u;
    g0.y = lds_base + (unsigned)(i * (CORNERS * ENC_DIM * 4));
    g0.z = (unsigned)(emb_a & 0xFFFFFFFFull);
    g0.w = (unsigned)((emb_a >> 32) & 0x01FFFFFFull) | 0x80000000u;

    // D# group 1: wg_mask=0, data_size=2(4B), tensor_dim0=32,
    //             tensor_dim1=2^19, tile_dim0=32, tile_dim1=8 (valid indices),
    //             tensor_dim0_stride=32 (ENC_DIM row pitch).
    i32x8 g1 = { 0x00020000,                 // [17:16] data_size = 4B
                 (ENC_DIM << 16),            // [63:48] tensor_dim0 lo = 32
                 0,                          // dim0 hi = 0, tensor_dim1 lo = 0
                 (ENC_DIM << 16) | 8,        // tensor_dim1 hi=8 (2^19), tile_dim0=32
                 CORNERS,                    // [143:128] tile_dim1 = 8 rows
                 ENC_DIM,                    // [191:160] tensor_dim0_stride = 32
                 0, 0 };

    i32x4 g2 = { t0, t1, t2, t3 };           // gather row indices 0..3 (32-bit)
    i32x4 g3 = { t4, t5, t6, t7 };           // gather row indices 4..7

    i32x8 gz = { 0, 0, 0, 0, 0, 0, 0, 0 };   // trailing group (unused; zero)

    __builtin_amdgcn_tensor_load_to_lds(g0, g1, g2, g3, gz, /*cpol*/0);
  }

  // ---------------- Phase 2: pipelined consume (in-order TENSORcnt) --------
  const size_t obase = (size_t)ray * (ENC_DEPTH * N_POINTS * ENC_DIM);
#pragma unroll
  for (int i = 0; i < ENC_DEPTH; ++i) {
    // Normalized coords for both points (register-only; overlaps TDM latency).
    float xd[N_POINTS], yd[N_POINTS], zd[N_POINTS];
#pragma unroll
    for (int p = 0; p < N_POINTS; ++p) {
      xd[p] = clamp01((px[p][0] - nmn[i][0]) * rex[i][0]);
      yd[p] = clamp01((px[p][1] - nmn[i][1]) * rex[i][1]);
      zd[p] = clamp01((px[p][2] - nmn[i][2]) * rex[i][2]);
    }

    // Depths complete in order: first i+1 gathers done once TENSORcnt<=7-i.
    wait_tensorcnt_le(ENC_DEPTH - 1 - i);

    // Corner order c: (0,0,0),(1,0,0),(0,1,0),(0,0,1),(1,0,1),(0,1,1),(1,1,0),(1,1,1)
    const float* f = &wbase[i * (CORNERS * ENC_DIM)];
    const float f0 = f[0 * ENC_DIM + lane];
    const float f1 = f[1 * ENC_DIM + lane];
    const float f2 = f[2 * ENC_DIM + lane];
    const float f3 = f[3 * ENC_DIM + lane];
    const float f4 = f[4 * ENC_DIM + lane];
    const float f5 = f[5 * ENC_DIM + lane];
    const float f6 = f[6 * ENC_DIM + lane];
    const float f7 = f[7 * ENC_DIM + lane];

#pragma unroll
    for (int p = 0; p < N_POINTS; ++p) {
      const float ax = 1.f - xd[p], ay = 1.f - yd[p], az = 1.f - zd[p];
      // Factored trilinear lerp tree (7 lerps == 14 VALU ops per point).
      const float c00 = fmaf(xd[p], f1, ax * f0);   // y=0, z=0
      const float c10 = fmaf(xd[p], f6, ax * f2);   // y=1, z=0
      const float c01 = fmaf(xd[p], f4, ax * f3);   // y=0, z=1
      const float c11 = fmaf(xd[p], f7, ax * f5);   // y=1, z=1
      const float cz0 = fmaf(yd[p], c10, ay * c00);
      const float cz1 = fmaf(yd[p], c11, ay * c01);
      const float acc = fmaf(zd[p], cz1, az * cz0);

      // Streaming output (128MB total): non-temporal so emb stays L2-resident.
      __builtin_nontemporal_store(
          acc, &out[obase + i * (N_POINTS * ENC_DIM) + p * ENC_DIM + lane]);
    }
  }
}

extern "C" void kernel_launch(void* const* d_in, const int* in_sizes, int n_in,
                              void* d_out, int out_size, void* d_ws, size_t ws_size,
                              hipStream_t stream) {
  const float* inp          = (const float*)d_in[0];
  const int*   history      = (const int*)  d_in[1];
  const float* emb          = (const float*)d_in[2];
  const float* nodes_min    = (const float*)d_in[3];
  const float* nodes_extent = (const float*)d_in[4];
  float*       out          = (float*)d_out;

  const int n_rays = in_sizes[0] / (N_POINTS * 3);
  const int blocks = (n_rays + WAVES_PER_BLOCK - 1) / WAVES_PER_BLOCK;
  nbvh_enc_kernel<<<blocks, WAVES_PER_BLOCK * 32, 0, stream>>>(
      inp, history, emb, nodes_min, nodes_extent, out, n_rays);
}